// wgnn_18047452578168
// MI455X (gfx1250) — compile-verified
//
#include <hip/hip_runtime.h>
#include <math.h>

#define NN      50000          // nodes
#define NE      800000         // edges (without self loops)
#define ET      (NE + NN)      // edges incl. self loops
#define CIN     128
#define COP     64
#define COUT    40
#define NLAYER  4
#define NT      (NN / 16)      // 3125 column tiles (exact)

typedef float v2f __attribute__((ext_vector_type(2)));
typedef float v8f __attribute__((ext_vector_type(8)));

__device__ __forceinline__ float lrelu02(float a) { return a > 0.f ? a : 0.2f * a; }

// float atomic max via monotonic int encoding (always compiles, matches FP order)
__device__ __forceinline__ void atomicMaxF(float* addr, float val) {
    if (val >= 0.f)
        atomicMax((int*)addr, __float_as_int(val));
    else
        atomicMin((unsigned int*)addr, (unsigned int)__float_as_int(val));
}

// -------- opening GEMM: X(64,NN) = relu(A(64,128) @ B(128,NN)), fp32 WMMA --------
__global__ __launch_bounds__(256) void k_open(const float* __restrict__ A,
                                              const float* __restrict__ B,
                                              float* __restrict__ X) {
    const int wave = (blockIdx.x * blockDim.x + threadIdx.x) >> 5;
    if (wave >= 4 * NT) return;
    const int lane = threadIdx.x & 31;
    const int l15  = lane & 15;
    const int hi   = lane >> 4;          // 0 | 1
    const int kh   = hi << 1;            // 0 | 2
    const int mB   = (wave / NT) * 16;
    const int nB   = (wave % NT) * 16;
    v8f acc = {};
    for (int k0 = 0; k0 < CIN; k0 += 4) {
        v2f a, b;
        a.x = A[(mB + l15) * CIN + k0 + kh];
        a.y = A[(mB + l15) * CIN + k0 + kh + 1];
        b.x = B[(size_t)(k0 + kh) * NN + nB + l15];
        b.y = B[(size_t)(k0 + kh + 1) * NN + nB + l15];
        acc = __builtin_amdgcn_wmma_f32_16x16x4_f32(false, a, false, b,
                                                    (short)0, acc, false, false);
    }
#pragma unroll
    for (int r = 0; r < 8; ++r) {
        const int row = mB + r + (hi << 3);
        X[(size_t)row * NN + nB + l15] = fmaxf(acc[r], 0.f);
    }
}

// -------- per-layer GEMM with fused omega-mix on the B operand --------
// Xout(64,NN) = relu( KN1(64,64) @ [ Xin + omega*(SpT^T - Xin) ] )
__global__ __launch_bounds__(256) void k_gemm_mix(const float* __restrict__ A,
                                                  const float* __restrict__ Xin,
                                                  const float* __restrict__ SpT,
                                                  const float* __restrict__ omg,
                                                  float* __restrict__ Xout) {
    const int wave = (blockIdx.x * blockDim.x + threadIdx.x) >> 5;
    if (wave >= 4 * NT) return;
    const int lane = threadIdx.x & 31;
    const int l15  = lane & 15;
    const int hi   = lane >> 4;
    const int kh   = hi << 1;
    const int mB   = (wave / NT) * 16;
    const int nB   = (wave % NT) * 16;
    const int n    = nB + l15;
    v8f acc = {};
    for (int k0 = 0; k0 < COP; k0 += 4) {
        const int ka = k0 + kh, kb = ka + 1;
        v2f a, b;
        a.x = A[(mB + l15) * COP + ka];
        a.y = A[(mB + l15) * COP + kb];
        const float x0 = Xin[(size_t)ka * NN + n];
        const float x1 = Xin[(size_t)kb * NN + n];
        b.x = x0 + omg[ka] * (SpT[(size_t)n * COP + ka] - x0);
        b.y = x1 + omg[kb] * (SpT[(size_t)n * COP + kb] - x1);
        acc = __builtin_amdgcn_wmma_f32_16x16x4_f32(false, a, false, b,
                                                    (short)0, acc, false, false);
    }
#pragma unroll
    for (int r = 0; r < 8; ++r) {
        const int row = mB + r + (hi << 3);
        Xout[(size_t)row * NN + nB + l15] = fmaxf(acc[r], 0.f);
    }
}

// -------- closing GEMM: OUT[n][m] = (KNclose(40,64) @ X)(m,n), transposed store --------
__global__ __launch_bounds__(256) void k_close(const float* __restrict__ A,
                                               const float* __restrict__ X,
                                               float* __restrict__ OUT) {
    const int wave = (blockIdx.x * blockDim.x + threadIdx.x) >> 5;
    if (wave >= 3 * NT) return;
    const int lane = threadIdx.x & 31;
    const int l15  = lane & 15;
    const int hi   = lane >> 4;
    const int kh   = hi << 1;
    const int mB   = (wave / NT) * 16;   // 0,16,32 (rows 40..47 padded)
    const int nB   = (wave % NT) * 16;
    const int arow = mB + l15;
    v8f acc = {};
    for (int k0 = 0; k0 < COP; k0 += 4) {
        v2f a, b;
        a.x = (arow < COUT) ? A[arow * COP + k0 + kh] : 0.f;
        a.y = (arow < COUT) ? A[arow * COP + k0 + kh + 1] : 0.f;
        b.x = X[(size_t)(k0 + kh) * NN + nB + l15];
        b.y = X[(size_t)(k0 + kh + 1) * NN + nB + l15];
        acc = __builtin_amdgcn_wmma_f32_16x16x4_f32(false, a, false, b,
                                                    (short)0, acc, false, false);
    }
#pragma unroll
    for (int r = 0; r < 8; ++r) {
        const int row = mB + r + (hi << 3);
        if (row < COUT)
            OUT[(size_t)(nB + l15) * COUT + row] = acc[r];
    }
}

// -------- per-node attention score prep + segment-state init --------
__global__ void k_scores(const float* __restrict__ X, const float* __restrict__ asrc,
                         const float* __restrict__ adst, float* __restrict__ ssrc,
                         float* __restrict__ sdst, float* __restrict__ mmax,
                         float* __restrict__ dsum) {
    const int n = blockIdx.x * blockDim.x + threadIdx.x;
    if (n >= NN) return;
    float s = 0.f, t = 0.f;
    for (int c = 0; c < COP; ++c) {
        const float v = X[(size_t)c * NN + n];
        s = fmaf(asrc[c], v, s);
        t = fmaf(adst[c], v, t);
    }
    ssrc[n] = s; sdst[n] = t;
    mmax[n] = -3.0e38f; dsum[n] = 0.f;
}

// -------- channel-major -> node-major transpose + zero scatter target --------
__global__ void k_transpose(const float* __restrict__ X, float* __restrict__ XT,
                            float* __restrict__ SpT) {
    const int idx = blockIdx.x * blockDim.x + threadIdx.x; // n*64 + c
    if (idx >= COP * NN) return;
    const int c = idx & (COP - 1);
    const int n = idx >> 6;
    XT[idx]  = X[(size_t)c * NN + n];
    SpT[idx] = 0.f;
}

// -------- edge pass 1: leaky-relu logits + segment max over destination I --------
__global__ void k_edge_max(const int* __restrict__ ei, const int* __restrict__ ej,
                           const float* __restrict__ ssrc, const float* __restrict__ sdst,
                           float* __restrict__ ew, float* __restrict__ mmax) {
    const int e = blockIdx.x * blockDim.x + threadIdx.x;
    if (e >= ET) return;
    int I, J;
    if (e < NE) { I = ei[e]; J = ej[e]; } else { I = J = e - NE; }
    const float a = lrelu02(ssrc[I] + sdst[J]);
    ew[e] = a;
    atomicMaxF(&mmax[I], a);
}

// -------- edge pass 2: exp(a - max) + segment sum --------
__global__ void k_edge_exp(const int* __restrict__ ei, float* __restrict__ ew,
                           const float* __restrict__ mmax, float* __restrict__ dsum) {
    const int e = blockIdx.x * blockDim.x + threadIdx.x;
    if (e >= ET) return;
    const int I = (e < NE) ? ei[e] : (e - NE);
    const float w = expf(ew[e] - mmax[I]);
    ew[e] = w;
    atomicAdd(&dsum[I], w);
}

// -------- edge pass 3: weighted gather of XT[J] -> atomic scatter into SpT[I] --------
// one wave per edge; lane owns 2 contiguous channels (node-major, 2 cachelines/edge)
__global__ __launch_bounds__(256) void k_scatter(const int* __restrict__ ei,
                                                 const int* __restrict__ ej,
                                                 const float* __restrict__ ew,
                                                 const float* __restrict__ dsum,
                                                 const float* __restrict__ XT,
                                                 float* __restrict__ SpT) {
    const int e = (blockIdx.x * blockDim.x + threadIdx.x) >> 5;
    if (e >= ET) return;
    const int lane = threadIdx.x & 31;
    int I, J;
    if (e < NE) { I = ei[e]; J = ej[e]; } else { I = J = e - NE; }
    const float coeff = ew[e] / dsum[I];
    const float2 f = *(const float2*)&XT[(size_t)J * COP + lane * 2];
    atomicAdd(&SpT[(size_t)I * COP + lane * 2 + 0], coeff * f.x);
    atomicAdd(&SpT[(size_t)I * COP + lane * 2 + 1], coeff * f.y);
}

// -------- per-node log-softmax over the 40 outputs (in place on d_out) --------
__global__ void k_logsoftmax(float* __restrict__ OUT) {
    const int n = blockIdx.x * blockDim.x + threadIdx.x;
    if (n >= NN) return;
    float v[COUT];
    float mx = -3.0e38f;
#pragma unroll
    for (int m = 0; m < COUT; ++m) { v[m] = OUT[(size_t)n * COUT + m]; mx = fmaxf(mx, v[m]); }
    float s = 0.f;
#pragma unroll
    for (int m = 0; m < COUT; ++m) s += expf(v[m] - mx);
    const float l = mx + logf(s);
#pragma unroll
    for (int m = 0; m < COUT; ++m) OUT[(size_t)n * COUT + m] = v[m] - l;
}

extern "C" void kernel_launch(void* const* d_in, const int* in_sizes, int n_in,
                              void* d_out, int out_size, void* d_ws, size_t ws_size,
                              hipStream_t stream) {
    const float* xn      = (const float*)d_in[0];
    const float* K1Nopen = (const float*)d_in[1];
    const float* KNclose = (const float*)d_in[2];
    const float* KN1     = (const float*)d_in[3];
    const float* att_src = (const float*)d_in[4];
    const float* att_dst = (const float*)d_in[5];
    const float* omega   = (const float*)d_in[6];
    const int*   ei      = (const int*)d_in[7];
    const int*   ej      = (const int*)d_in[8];
    float* out = (float*)d_out;

    // workspace layout (floats): 2 feature buffers + transposed copy + scatter
    // target + per-node segment state + per-edge weights  (~55 MB total)
    const size_t CN = (size_t)COP * NN;
    float* xA   = (float*)d_ws;
    float* xB   = xA + CN;
    float* XT   = xB + CN;
    float* SpT  = XT + CN;
    float* ssrc = SpT + CN;
    float* sdst = ssrc + NN;
    float* mmax = sdst + NN;
    float* dsum = mmax + NN;
    float* ew   = dsum + NN;

    k_open<<<(4 * NT + 7) / 8, 256, 0, stream>>>(K1Nopen, xn, xA);

    float* cur = xA;
    float* nxt = xB;
    for (int i = 0; i < NLAYER; ++i) {
        k_scores<<<(NN + 255) / 256, 256, 0, stream>>>(cur, att_src + i * COP,
                                                       att_dst + i * COP,
                                                       ssrc, sdst, mmax, dsum);
        k_transpose<<<(COP * NN + 255) / 256, 256, 0, stream>>>(cur, XT, SpT);
        k_edge_max<<<(ET + 255) / 256, 256, 0, stream>>>(ei, ej, ssrc, sdst, ew, mmax);
        k_edge_exp<<<(ET + 255) / 256, 256, 0, stream>>>(ei, ew, mmax, dsum);
        k_scatter<<<(ET + 7) / 8, 256, 0, stream>>>(ei, ej, ew, dsum, XT, SpT);
        k_gemm_mix<<<(4 * NT + 7) / 8, 256, 0, stream>>>(KN1 + (size_t)i * COP * COP,
                                                         cur, SpT, omega + i * COP, nxt);
        float* t = cur; cur = nxt; nxt = t;
    }

    k_close<<<(3 * NT + 7) / 8, 256, 0, stream>>>(KNclose, cur, out);
    k_logsoftmax<<<(NN + 255) / 256, 256, 0, stream>>>(out);
}